// DetectionProposal_36807869727183
// MI455X (gfx1250) — compile-verified
//
#include <hip/hip_runtime.h>
#include <stdint.h>

// Problem constants (from reference setup_inputs)
#define B_   8
#define N_   100000
#define C_   10
#define K1_  1000
#define KP_  1024          // padded per-class candidate count (pow2 for bitonic)
#define CK_  (C_ * K1_)    // 10000
#define CKP_ 16384         // padded merge count
#define TPB  256           // 8 wave32s per block
#define MIN_CONF_ 0.05f
#define NMS_IOU_  0.4f
#define POST_IOU_ 0.65f

// top-k selection machinery
#define SLICES_ 8          // anchor slices per batch for contiguous sweeps
#define HBITS_  12
#define HBINS_  4096       // top-12-bit float-key histogram
#define CAP_    8192       // per-(b,c) candidate cap (pow2 for bitonic)

typedef float v8f_ __attribute__((ext_vector_type(8)));
typedef float v2f_ __attribute__((ext_vector_type(2)));
typedef unsigned long long u64_;

#if defined(__has_builtin)
# if __has_builtin(__builtin_amdgcn_global_load_async_to_lds_b64) && \
     __has_builtin(__builtin_amdgcn_s_wait_asynccnt)
#  define HAVE_ASYNC_LDS 1
# endif
#endif
#ifndef HAVE_ASYNC_LDS
# define HAVE_ASYNC_LDS 0
#endif

#if HAVE_ASYNC_LDS
// builtin signature (from compiler diagnostic): takes pointers to
// __vector_size__(8) int in global (AS1) / LDS (AS3) address spaces.
typedef int v2i_vs __attribute__((vector_size(2 * sizeof(int))));
typedef __attribute__((address_space(1))) v2i_vs* async_gptr_t;
typedef __attribute__((address_space(3))) v2i_vs* async_lptr_t;
#endif

// ---------------------------------------------------------------------------
// zero scratch counters (hist + cnt are contiguous in ws)
// ---------------------------------------------------------------------------
__global__ void __launch_bounds__(TPB) zero_u32(unsigned int* __restrict__ p, int n)
{
  int i = blockIdx.x * TPB + threadIdx.x;
  if (i < n) p[i] = 0u;
}

// ---------------------------------------------------------------------------
// Pass 1: contiguous sweep of (N,C)-interleaved scores; 10 class histograms
// (top-12 float-key bits) in 160KB LDS, merged to global.
// ---------------------------------------------------------------------------
__global__ void __launch_bounds__(TPB) class_hist(
    const float* __restrict__ cls_pred, unsigned int* __restrict__ hist)
{
  extern __shared__ char smem[];
  unsigned int* lh = (unsigned int*)smem;              // C_*HBINS_
  const int tid = threadIdx.x;
  const int b = blockIdx.x / SLICES_;
  const int sl = blockIdx.x % SLICES_;
  for (int i = tid; i < C_ * HBINS_; i += TPB) lh[i] = 0u;
  __syncthreads();

  const int per = (N_ + SLICES_ - 1) / SLICES_;        // 12500 anchors
  const int f0 = sl * per * C_;
  int f1 = (sl + 1) * per * C_;
  if (f1 > N_ * C_) f1 = N_ * C_;
  const float* base = cls_pred + (size_t)b * N_ * C_;
  for (int f = f0 + tid; f < f1; f += TPB) {
    if (f + TPB < f1)
      __builtin_prefetch(&base[f + TPB], 0, 1);        // global_prefetch_b8
    float s = base[f];
    unsigned int key = (s >= MIN_CONF_) ? __float_as_uint(s) : 0u;
    int c = f % C_;
    atomicAdd(&lh[c * HBINS_ + (key >> (32 - HBITS_))], 1u);
  }
  __syncthreads();

  unsigned int* gh = hist + (size_t)b * C_ * HBINS_;
  for (int i = tid; i < C_ * HBINS_; i += TPB) {
    unsigned int v = lh[i];
    if (v) atomicAdd(&gh[i], v);
  }
}

// ---------------------------------------------------------------------------
// Pass 2: per (b,c) suffix-sum the 4096 bins, pick the bin where the
// from-the-top cumulative count first reaches K1 -> key threshold.
// ---------------------------------------------------------------------------
__global__ void __launch_bounds__(TPB) find_pivot(
    const unsigned int* __restrict__ hist, unsigned int* __restrict__ thr)
{
  __shared__ unsigned int buf[HBINS_];
  const int tid = threadIdx.x;
  const unsigned int* gh = hist + (size_t)blockIdx.x * HBINS_;
  for (int i = tid; i < HBINS_; i += TPB) buf[i] = gh[i];
  __syncthreads();
  for (int d = 1; d < HBINS_; d <<= 1) {               // inclusive suffix sum
    unsigned int tmp[HBINS_ / TPB];
    for (int i = tid, k = 0; i < HBINS_; i += TPB, ++k)
      tmp[k] = (i + d < HBINS_) ? buf[i + d] : 0u;
    __syncthreads();
    for (int i = tid, k = 0; i < HBINS_; i += TPB, ++k)
      buf[i] += tmp[k];
    __syncthreads();
  }
  if (tid == 0 && buf[0] < K1_) thr[blockIdx.x] = 0u;  // < K1 valid entries
  for (int i = tid; i < HBINS_; i += TPB) {
    bool piv = (buf[i] >= (unsigned int)K1_) &&
               (i == HBINS_ - 1 || buf[i + 1] < (unsigned int)K1_);
    if (piv) thr[blockIdx.x] = (unsigned int)i << (32 - HBITS_);
  }
}

// ---------------------------------------------------------------------------
// Pass 3: second contiguous sweep (L2-resident); scatter candidates >= thr.
// ---------------------------------------------------------------------------
__global__ void __launch_bounds__(TPB) collect_cands(
    const float* __restrict__ cls_pred, const unsigned int* __restrict__ thr,
    unsigned int* __restrict__ cnt, u64_* __restrict__ cand)
{
  __shared__ unsigned int thrs[C_];
  const int tid = threadIdx.x;
  const int b = blockIdx.x / SLICES_;
  const int sl = blockIdx.x % SLICES_;
  if (tid < C_) thrs[tid] = thr[b * C_ + tid];
  __syncthreads();

  const int per = (N_ + SLICES_ - 1) / SLICES_;
  const int f0 = sl * per * C_;
  int f1 = (sl + 1) * per * C_;
  if (f1 > N_ * C_) f1 = N_ * C_;
  const float* base = cls_pred + (size_t)b * N_ * C_;
  for (int f = f0 + tid; f < f1; f += TPB) {
    if (f + TPB < f1)
      __builtin_prefetch(&base[f + TPB], 0, 1);
    float s = base[f];
    unsigned int key = (s >= MIN_CONF_) ? __float_as_uint(s) : 0u;
    int i = f / C_;
    int c = f - i * C_;
    if (key != 0u && key >= thrs[c]) {
      int bc = b * C_ + c;
      unsigned int pos = atomicAdd(&cnt[bc], 1u);
      if (pos < CAP_)
        cand[(size_t)bc * CAP_ + pos] =
            ((u64_)key << 32) | (u64_)(~(unsigned int)i);
    }
  }
}

// ---------------------------------------------------------------------------
// Pass 4: per (b,c) bitonic sort of <=8192 candidates (desc, lowest-index
// tie-break => deterministic & matches jax top_k); emit sorted top-K1 with
// corner boxes. Candidate staging uses ASYNC-to-LDS when available.
// ---------------------------------------------------------------------------
__global__ void __launch_bounds__(TPB) sort_emit(
    const u64_* __restrict__ cand, const unsigned int* __restrict__ cnt,
    const float* __restrict__ boxes,
    float* __restrict__ o_score, int* __restrict__ o_idx,
    float* __restrict__ o_corners)
{
  extern __shared__ char smem[];
  u64_* list = (u64_*)smem;                            // CAP_ entries (64 KB)
  const int tid = threadIdx.x;
  const int bc = blockIdx.x;
  const int b = bc / C_;
  int total = (int)cnt[bc]; if (total > CAP_) total = CAP_;
  const u64_* src = cand + (size_t)bc * CAP_;

#if HAVE_ASYNC_LDS
  for (int i = tid; i < total; i += TPB) {
    __builtin_amdgcn_global_load_async_to_lds_b64(
        (async_gptr_t)(u64_*)(src + i), (async_lptr_t)(list + i), 0, 0);
  }
  __builtin_amdgcn_s_wait_asynccnt(0);
#else
  for (int i = tid; i < total; i += TPB) list[i] = src[i];
#endif
  __syncthreads();
  for (int i = total + tid; i < CAP_; i += TPB)
    list[i] = 0x00000000FFFFFFFFull;                   // key=0, idx decodes 0
  __syncthreads();

  for (int k = 2; k <= CAP_; k <<= 1) {
    for (int j = k >> 1; j > 0; j >>= 1) {
      for (int i = tid; i < CAP_; i += TPB) {
        int ixj = i ^ j;
        if (ixj > i) {
          u64_ a = list[i], bb = list[ixj];
          bool desc = ((i & k) == 0);
          if (desc ? (a < bb) : (a > bb)) { list[i] = bb; list[ixj] = a; }
        }
      }
      __syncthreads();
    }
  }

  const float* boxb = boxes + (size_t)b * N_ * 4;
  for (int r = tid; r < K1_; r += TPB) {
    u64_ x = list[r];
    unsigned int key = (unsigned int)(x >> 32);
    unsigned int idx = ~(unsigned int)x;
    float score = key ? __uint_as_float(key) : -1.0f;
    size_t slot = (size_t)bc * K1_ + r;
    o_score[slot] = score;
    o_idx[slot] = (int)idx;
    float cx = boxb[(size_t)idx * 4 + 0];
    float cy = boxb[(size_t)idx * 4 + 1];
    float w  = boxb[(size_t)idx * 4 + 2];
    float h  = boxb[(size_t)idx * 4 + 3];
    o_corners[slot * 4 + 0] = cy - 0.5f * h;
    o_corners[slot * 4 + 1] = cx - 0.5f * w;
    o_corners[slot * 4 + 2] = cy + 0.5f * h;
    o_corners[slot * 4 + 3] = cx + 0.5f * w;
  }
}

// ---------------------------------------------------------------------------
// NMS core: 160KB-LDS resident. Suppression bitmask built in 16x16 tiles; the
// bilinear union term (area_i + area_j) comes from V_WMMA_F32_16X16X4_F32,
// intersections from VALU in the matching C/D lane layout, __ballot packs the
// predicate tile into row-major mask words. Greedy scan by one wave32.
// ---------------------------------------------------------------------------
struct NmsShared {
  float* y1; float* x1; float* y2; float* x2; float* area;
  int* valid; int* keep; int* pos;
  unsigned int* supp; unsigned int* mask;
};

__device__ __forceinline__ NmsShared nms_layout(char* smem) {
  NmsShared s;
  float* f = (float*)smem;
  s.y1 = f;            s.x1 = f + KP_;      s.y2 = f + 2 * KP_;
  s.x2 = f + 3 * KP_;  s.area = f + 4 * KP_;
  s.valid = (int*)(f + 5 * KP_);
  s.keep  = s.valid + KP_;
  s.pos   = s.keep + KP_;
  s.supp  = (unsigned int*)(s.pos + KP_);
  s.mask  = s.supp + 32;
  return s;
}
#define NMS_SHM_BYTES ((size_t)(8 * KP_ * 4 + 32 * 4 + KP_ * 32 * 4))  // ~160 KB

__device__ __forceinline__ void iou_tile_wmma(const NmsShared& s, int ti, int tj,
                                              float thresh, int lane)
{
  const int half = lane >> 4;        // 0: lanes 0-15, 1: lanes 16-31
  const int l = lane & 15;
  // A (16x4 f32): row m = [area_m, 1, 0, 0]; B (4x16): col n = [1; area_n; 0; 0]
  v2f_ a, bv;
  if (half == 0) { a.x = s.area[ti * 16 + l]; a.y = 1.0f;
                   bv.x = 1.0f;               bv.y = s.area[tj * 16 + l]; }
  else           { a.x = 0.0f; a.y = 0.0f; bv.x = 0.0f; bv.y = 0.0f; }
  v8f_ acc = {};
  // D[m][n] = area[ti*16+m] + area[tj*16+n]
  acc = __builtin_amdgcn_wmma_f32_16x16x4_f32(false, a, false, bv,
                                              (short)0, acc, false, false);
  const int n = tj * 16 + l;
  const float cy1 = s.y1[n], cx1 = s.x1[n], cy2 = s.y2[n], cx2 = s.x2[n];
#pragma unroll
  for (int v = 0; v < 8; ++v) {
    const int m = ti * 16 + v + half * 8;   // matches C/D VGPR layout
    float ih = fminf(s.y2[m], cy2) - fmaxf(s.y1[m], cy1);
    float iw = fminf(s.x2[m], cx2) - fmaxf(s.x1[m], cx1);
    float inter = fmaxf(ih, 0.0f) * fmaxf(iw, 0.0f);
    float uni = fmaxf(acc[v] - inter, 1e-8f);
    bool hit = inter > thresh * uni;        // iou > thresh
    unsigned int w = (unsigned int)__ballot(hit);  // wave32: low 32 bits
    if (lane == 0) {
      const int word = tj >> 1;
      const int sh = (tj & 1) * 16;
      atomicOr(&s.mask[(ti * 16 + v) * 32 + word],     (w & 0xFFFFu) << sh);
      atomicOr(&s.mask[(ti * 16 + v + 8) * 32 + word], (w >> 16) << sh);
    }
  }
}

__device__ void nms_core(char* smem, const float* __restrict__ scores,
                         const float* __restrict__ corners, float thresh)
{
  NmsShared s = nms_layout(smem);
  const int tid = threadIdx.x;
  for (int i = tid; i < KP_; i += TPB) {
    if (i < K1_) {
      float sc = scores[i];
      s.valid[i] = (sc >= MIN_CONF_) ? 1 : 0;
      float y1 = corners[i * 4 + 0], x1 = corners[i * 4 + 1];
      float y2 = corners[i * 4 + 2], x2 = corners[i * 4 + 3];
      s.y1[i] = y1; s.x1[i] = x1; s.y2[i] = y2; s.x2[i] = x2;
      s.area[i] = (y2 - y1) * (x2 - x1);
    } else {
      s.valid[i] = 0;
      s.y1[i] = 0.f; s.x1[i] = 0.f; s.y2[i] = 0.f; s.x2[i] = 0.f;
      s.area[i] = 0.f;
    }
    s.keep[i] = 0;
  }
  for (int i = tid; i < KP_ * 32; i += TPB) s.mask[i] = 0u;
  if (tid < 32) s.supp[tid] = 0u;
  __syncthreads();

  // upper-triangular 16x16 tiles; 2080 tiles / 8 waves = 260 each (uniform)
  const int T = KP_ / 16;                // 64
  const int NT = T * (T + 1) / 2;        // 2080
  const int wave = tid >> 5, lane = tid & 31;
  for (int t = wave; t < NT; t += TPB / 32) {
    int ti = 0, r = t;
    while (r >= (T - ti)) { r -= (T - ti); ++ti; }
    iou_tile_wmma(s, ti, ti + r, thresh, lane);
  }
  __syncthreads();

  // greedy scan: single wave32, 32 suppression words in lockstep
  if (tid < 32) {
    volatile unsigned int* supp = s.supp;
    for (int i = 0; i < K1_; ++i) {
      unsigned int sw = supp[i >> 5];
      bool active = s.valid[i] && (((sw >> (i & 31)) & 1u) == 0u);
      if (active) supp[lane] |= s.mask[i * 32 + lane];
      if (lane == 0) s.keep[i] = active ? 1 : 0;
    }
  }
  __syncthreads();
}

// ---------------------------------------------------------------------------
// per-class NMS @ 0.4; suppressed entries' scores -> -1 (in place)
// ---------------------------------------------------------------------------
__global__ void __launch_bounds__(TPB) nms_class(float* __restrict__ scores,
                                                 const float* __restrict__ corners)
{
  extern __shared__ char smem[];
  const size_t base = (size_t)blockIdx.x * K1_;
  nms_core(smem, scores + base, corners + base * 4, NMS_IOU_);
  NmsShared s = nms_layout(smem);
  for (int i = threadIdx.x; i < K1_; i += TPB)
    if (!s.keep[i]) scores[base + i] = -1.0f;
}

// ---------------------------------------------------------------------------
// per-batch merge of C*K1 candidates, bitonic top-K1 (desc, stable)
// ---------------------------------------------------------------------------
__global__ void __launch_bounds__(TPB) merge_topk(
    const float* __restrict__ scores, const int* __restrict__ idx,
    const float* __restrict__ corners,
    float* __restrict__ o_score, int* __restrict__ o_cls,
    int* __restrict__ o_oidx, float* __restrict__ o_corners)
{
  extern __shared__ char smem[];
  u64_* list = (u64_*)smem;
  const int tid = threadIdx.x;
  const int b = blockIdx.x;
  const size_t base = (size_t)b * CK_;
  for (int i = tid; i < CKP_; i += TPB) {
    if (i < CK_) {
      float sc = scores[base + i];
      unsigned int key = (sc >= MIN_CONF_) ? __float_as_uint(sc) : 0u;
      list[i] = ((u64_)key << 32) | (u64_)(~(unsigned int)i);
    } else {
      list[i] = 0x00000000FFFFFFFFull;
    }
  }
  __syncthreads();
  for (int k = 2; k <= CKP_; k <<= 1) {
    for (int j = k >> 1; j > 0; j >>= 1) {
      for (int i = tid; i < CKP_; i += TPB) {
        int ixj = i ^ j;
        if (ixj > i) {
          u64_ a = list[i], bb = list[ixj];
          bool desc = ((i & k) == 0);
          if (desc ? (a < bb) : (a > bb)) { list[i] = bb; list[ixj] = a; }
        }
      }
      __syncthreads();
    }
  }
  for (int r = tid; r < K1_; r += TPB) {
    u64_ x = list[r];
    unsigned int key = (unsigned int)(x >> 32);
    unsigned int slot = ~(unsigned int)x;   // pad decodes to 0 (row invalid anyway)
    size_t o = (size_t)b * K1_ + r;
    o_score[o] = key ? __uint_as_float(key) : -1.0f;
    o_cls[o]  = (int)(slot / K1_);
    o_oidx[o] = idx[base + slot];
    o_corners[o * 4 + 0] = corners[(base + slot) * 4 + 0];
    o_corners[o * 4 + 1] = corners[(base + slot) * 4 + 1];
    o_corners[o * 4 + 2] = corners[(base + slot) * 4 + 2];
    o_corners[o * 4 + 3] = corners[(base + slot) * 4 + 3];
  }
}

// ---------------------------------------------------------------------------
// final NMS @ 0.65, stable compaction of kept rows, emit (B,1000,6)
// ---------------------------------------------------------------------------
__global__ void __launch_bounds__(TPB) final_nms_emit(
    const float* __restrict__ s2_score, const int* __restrict__ s2_cls,
    const int* __restrict__ s2_oidx, const float* __restrict__ s2_corners,
    const float* __restrict__ boxes, float* __restrict__ out)
{
  extern __shared__ char smem[];
  const int tid = threadIdx.x;
  const int b = blockIdx.x;
  const size_t base = (size_t)b * K1_;
  nms_core(smem, s2_score + base, s2_corners + base * 4, POST_IOU_);
  NmsShared s = nms_layout(smem);
  if (tid == 0) {
    int p = 0;
    for (int i = 0; i < K1_; ++i) { s.pos[i] = p; p += s.keep[i]; }
  }
  float* ob = out + (size_t)b * K1_ * 6;
  for (int i = tid; i < K1_ * 6; i += TPB) ob[i] = 0.0f;
  __syncthreads();
  for (int r = tid; r < K1_; r += TPB) {
    if (s.keep[r]) {
      int p = s.pos[r];
      int oi = s2_oidx[base + r];
      const float* bx = boxes + ((size_t)b * N_ + (size_t)oi) * 4;
      ob[p * 6 + 0] = bx[0];
      ob[p * 6 + 1] = bx[1];
      ob[p * 6 + 2] = bx[2];
      ob[p * 6 + 3] = bx[3];
      ob[p * 6 + 4] = (float)s2_cls[base + r];
      ob[p * 6 + 5] = s2_score[base + r];
    }
  }
}

// ---------------------------------------------------------------------------
extern "C" void kernel_launch(void* const* d_in, const int* in_sizes, int n_in,
                              void* d_out, int out_size, void* d_ws, size_t ws_size,
                              hipStream_t stream) {
  (void)in_sizes; (void)n_in; (void)out_size; (void)ws_size;
  const float* cls_pred = (const float*)d_in[0];   // (B,N,C) f32
  const float* boxes    = (const float*)d_in[1];   // (B,N,4) f32
  float* out = (float*)d_out;                      // (B,1000,6) f32

  char* ws = (char*)d_ws;
  size_t off = 0;
  // hist + cnt contiguous so one zero pass clears both
  unsigned int* hist = (unsigned int*)(ws + off); off += (size_t)B_ * C_ * HBINS_ * 4;
  unsigned int* cnt  = (unsigned int*)(ws + off); off += (size_t)B_ * C_ * 4;
  unsigned int* thr  = (unsigned int*)(ws + off); off += (size_t)B_ * C_ * 4;
  u64_*  cand     = (u64_*) (ws + off); off += (size_t)B_ * C_ * CAP_ * 8;
  float* ws_score = (float*)(ws + off); off += (size_t)B_ * CK_ * 4;
  int*   ws_idx   = (int*)  (ws + off); off += (size_t)B_ * CK_ * 4;
  float* ws_corn  = (float*)(ws + off); off += (size_t)B_ * CK_ * 16;
  float* s2_score = (float*)(ws + off); off += (size_t)B_ * K1_ * 4;
  int*   s2_cls   = (int*)  (ws + off); off += (size_t)B_ * K1_ * 4;
  int*   s2_oidx  = (int*)  (ws + off); off += (size_t)B_ * K1_ * 4;
  float* s2_corn  = (float*)(ws + off); off += (size_t)B_ * K1_ * 16;

  const int nzero = B_ * C_ * HBINS_ + B_ * C_;    // hist + cnt
  zero_u32<<<(nzero + TPB - 1) / TPB, TPB, 0, stream>>>(hist, nzero);

  class_hist<<<B_ * SLICES_, TPB, (size_t)C_ * HBINS_ * 4, stream>>>(cls_pred, hist);
  find_pivot<<<B_ * C_, TPB, 0, stream>>>(hist, thr);
  collect_cands<<<B_ * SLICES_, TPB, 0, stream>>>(cls_pred, thr, cnt, cand);
  sort_emit<<<B_ * C_, TPB, (size_t)CAP_ * 8, stream>>>(cand, cnt, boxes,
                                                        ws_score, ws_idx, ws_corn);

  nms_class<<<B_ * C_, TPB, NMS_SHM_BYTES, stream>>>(ws_score, ws_corn);
  merge_topk<<<B_, TPB, (size_t)CKP_ * 8, stream>>>(ws_score, ws_idx, ws_corn,
                                                    s2_score, s2_cls, s2_oidx,
                                                    s2_corn);
  final_nms_emit<<<B_, TPB, NMS_SHM_BYTES, stream>>>(s2_score, s2_cls, s2_oidx,
                                                     s2_corn, boxes, out);
}